// network_70832600646077
// MI455X (gfx1250) — compile-verified
//
#include <hip/hip_runtime.h>
#include <math.h>

// ---------------------------------------------------------------------------
// MI455X (gfx1250) implementation.
//  - All big tensors kept in reference [B,C,N] layout (L==1).
//  - Six 1x1-conv GEMMs + kv/attn einsums run on V_WMMA_F32_16X16X4_F32.
//  - Conv input tiles staged to LDS (once per block, shared by 8 waves),
//    preferably via the Tensor Data Mover (tensor_load_to_lds + TENSORcnt).
//  - attn_dyn == value exactly (softmax row-sums are 1) -> Aapt folded away.
//  - Workspace use: ~102 MB (3 ping-pong tensors of 33.5 MB + small areas).
// ---------------------------------------------------------------------------

#define Bb 32
#define Cc 128
#define Nn 2048
#define Hh 8
#define DKk 16
#define LN_EPS 1e-5f

typedef __attribute__((ext_vector_type(2))) float v2f;
typedef __attribute__((ext_vector_type(8))) float v8f;
typedef __attribute__((ext_vector_type(4))) unsigned int v4u;
typedef __attribute__((ext_vector_type(4))) int v4i;
typedef __attribute__((ext_vector_type(8))) int v8i;

#if defined(__has_builtin)
#if __has_builtin(__builtin_amdgcn_tensor_load_to_lds) && \
    __has_builtin(__builtin_amdgcn_s_wait_tensorcnt)
#define USE_TDM 1
#endif
#endif
#ifndef USE_TDM
#define USE_TDM 0
#endif

// D(16x16,f32) = A(16x4,f32) * B(4x16,f32) + C
// A frag: lane<16 -> row M=lane, v0:K=0, v1:K=1 ; lane>=16 -> v0:K=2, v1:K=3
// (per CDNA5 ISA 7.12.2 "32-bit A-Matrix 16x4"); B mirrored.
// D frag: VGPR g, lanes0-15 -> (M=g, N=lane); lanes16-31 -> (M=g+8, N=lane-16)
__device__ __forceinline__ v8f wmma4(v2f a, v2f b, v8f c) {
  return __builtin_amdgcn_wmma_f32_16x16x4_f32(
      /*neg_a=*/false, a, /*neg_b=*/false, b,
      /*c_mod=*/(short)0, c, /*reuse_a=*/false, /*reuse_b=*/false);
}

#define ZERO8 {0.f, 0.f, 0.f, 0.f, 0.f, 0.f, 0.f, 0.f}

// ---------------------------------------------------------------------------
// Stage a [C=128 rows][16 cols] f32 tile (tile start Xb + n0, row stride Nn)
// into LDS as xs[c*16 + p].  TDM path: one DMA descriptor issued by wave 0,
// completion via TENSORcnt; fallback: cooperative vector loads.
__device__ __forceinline__ void stage_tile(const float* __restrict__ Xb,
                                           int n0, float* __restrict__ xs) {
#if USE_TDM
  if (threadIdx.x < 32) {  // one wave issues the DMA (EXEC ignored by TDM)
    const unsigned lds_off = (unsigned)(size_t)(void*)xs;  // LDS aperture low32
    const unsigned long long ga = (unsigned long long)(size_t)(Xb + n0);
    // D# group0: count=1 | lds_addr | global_addr[56:0] | type=2
    const v4u g0 = {1u, lds_off, (unsigned)ga,
                    (unsigned)((ga >> 32) & 0x01FFFFFFu) | 0x80000000u};
    // D# group1: data_size=4B; tensor_dim0=16, tensor_dim1=128;
    //            tile_dim0=16, tile_dim1=128; dim0_stride=Nn=2048
    const v8i g1 = {(int)0x00020000, (int)(16u << 16), (int)(128u << 16),
                    (int)(16u << 16), 128, Nn, 0, 0};
    const v4i gz = {0, 0, 0, 0};
#if __clang_major__ >= 23
    const v8i g1z = {0, 0, 0, 0, 0, 0, 0, 0};
    __builtin_amdgcn_tensor_load_to_lds(g0, g1, gz, gz, g1z, 0);
#else
    __builtin_amdgcn_tensor_load_to_lds(g0, g1, gz, gz, 0);
#endif
    __builtin_amdgcn_s_wait_tensorcnt(0);
  }
#else
  const int t = threadIdx.x;
#pragma unroll
  for (int i = 0; i < 8; ++i) {
    const int idx = i * 256 + t;  // 0..2047
    xs[idx] = Xb[(size_t)(idx >> 4) * Nn + n0 + (idx & 15)];
  }
#endif
  __syncthreads();
}

// ---------------------------------------------------------------------------
// K0: transpose the six [C,C] weight matrices: Wt[c][o] = W[o][c]
__global__ __launch_bounds__(256) void transpose_w_kernel(
    const float* __restrict__ W0, const float* __restrict__ W1,
    const float* __restrict__ W2, const float* __restrict__ W3,
    const float* __restrict__ W4, const float* __restrict__ W5,
    float* __restrict__ Wt) {
  const float* Ws[6] = {W0, W1, W2, W3, W4, W5};
  const int m = blockIdx.y;
  const float* W = Ws[m];
  const int idx = blockIdx.x * 256 + threadIdx.x;  // 0..16383
  const int o = idx >> 7;
  const int c = idx & 127;
  Wt[m * (Cc * Cc) + c * Cc + o] = W[idx];
}

// ---------------------------------------------------------------------------
// K1: key = softmax(memory / 4) over DK.  memory [H,1,N,DK] -> Ksm [H*N,16]
__global__ __launch_bounds__(256) void key_softmax_kernel(
    const float* __restrict__ Mem, float* __restrict__ Ksm) {
  const int r = blockIdx.x * 256 + threadIdx.x;  // h*N + n
  const float* p = Mem + (size_t)r * DKk;
  float v[DKk];
  float mx = -INFINITY;
#pragma unroll
  for (int i = 0; i < DKk; ++i) { v[i] = p[i] * 0.25f; mx = fmaxf(mx, v[i]); }
  float s = 0.f;
#pragma unroll
  for (int i = 0; i < DKk; ++i) { v[i] = expf(v[i] - mx); s += v[i]; }
  const float inv = 1.f / s;
  float* o = Ksm + (size_t)r * DKk;
#pragma unroll
  for (int i = 0; i < DKk; ++i) o[i] = v[i] * inv;
}

// ---------------------------------------------------------------------------
// Shared WMMA k-loop: D[o-tile 16][n-tile 16] over K=C=128.
// xs: LDS-staged X tile [128][16]; Wat/Wbt: pre-transposed weights Wt[c][o].
template <bool DUAL>
__device__ __forceinline__ void conv_kloop(
    const float* __restrict__ xs, const float* __restrict__ Wat,
    const float* __restrict__ Wbt, int otile, int half, int col, v8f& acc0,
    v8f& acc1) {
#pragma unroll 4
  for (int k0 = 0; k0 < Cc; k0 += 4) {
    const int ka = k0 + 2 * half;
    v2f bf;                                  // B[k][p] = xs[ka..ka+1][col]
    bf.x = xs[(ka + 0) * 16 + col];
    bf.y = xs[(ka + 1) * 16 + col];
    v2f a0;                                  // A[m][k] = Wt[ka..ka+1][otile+col]
    a0.x = Wat[(ka + 0) * Cc + otile + col];
    a0.y = Wat[(ka + 1) * Cc + otile + col];
    acc0 = wmma4(a0, bf, acc0);
    if (DUAL) {
      v2f a1;
      a1.x = Wbt[(ka + 0) * Cc + otile + col];
      a1.y = Wbt[(ka + 1) * Cc + otile + col];
      acc1 = wmma4(a1, bf, acc1);
    }
  }
}

// Prefetch the next n-tile of the input (first-touch HBM latency hiding).
__device__ __forceinline__ void prefetch_next_tile(const float* __restrict__ Xb,
                                                   int n0) {
  const int t = threadIdx.x;
  if (n0 + 16 < Nn)
    __builtin_prefetch(&Xb[(size_t)(t & 127) * Nn + n0 + 16 + ((t >> 7) << 3)],
                       0, 1);
}

// ---------------------------------------------------------------------------
// K2: q = relu(Wq x + bq), v = relu(Wv x + bv); query = softmax(q/4, over DK).
// Each wave owns one head's 16 channels (otile = wave*16 == head boundary):
// the DK-softmax is over the tile's M dim: 8 regs in-lane + shfl_xor(16).
// grid (N/16, B), block 256 (8 waves)
__global__ __launch_bounds__(256) void conv_qv_kernel(
    const float* __restrict__ X, const float* __restrict__ Wqt,
    const float* __restrict__ Wvt, const float* __restrict__ bq,
    const float* __restrict__ bv, float* __restrict__ Q,
    float* __restrict__ V) {
  __shared__ float xs[Cc * 16];
  const int lane = threadIdx.x & 31;
  const int wave = threadIdx.x >> 5;
  const int half = lane >> 4;
  const int col = lane & 15;
  const int n0 = blockIdx.x * 16;
  const int b = blockIdx.y;
  const int otile = wave * 16;  // == head * DK

  const float* Xb = X + (size_t)b * Cc * Nn;
  prefetch_next_tile(Xb, n0);
  stage_tile(Xb, n0, xs);

  v8f accq = ZERO8, accv = ZERO8;
  conv_kloop<true>(xs, Wqt, Wvt, otile, half, col, accq, accv);

  float qv[8], mx = -INFINITY;
#pragma unroll
  for (int g = 0; g < 8; ++g) {
    const int o = otile + g + 8 * half;
    float t = fmaxf(accq[g] + bq[o], 0.f) * 0.25f;  // relu then /sqrt(DK)
    qv[g] = t;
    mx = fmaxf(mx, t);
  }
  mx = fmaxf(mx, __shfl_xor(mx, 16, 32));  // across M halves
  float s = 0.f;
#pragma unroll
  for (int g = 0; g < 8; ++g) { qv[g] = expf(qv[g] - mx); s += qv[g]; }
  s += __shfl_xor(s, 16, 32);
  const float inv = 1.f / s;
#pragma unroll
  for (int g = 0; g < 8; ++g) {
    const int o = otile + g + 8 * half;
    const size_t idx = (size_t)b * Cc * Nn + (size_t)o * Nn + n0 + col;
    Q[idx] = qv[g] * inv;
    V[idx] = fmaxf(accv[g] + bv[o], 0.f);
  }
}

// ---------------------------------------------------------------------------
// K3: kvT[b,h][y][x] = sum_n V[b, h*16+y, n] * key[h, n, x]   (K = N = 2048)
// 8 waves split K; LDS tree-combine of the 8 partial 16x16 tiles.
// grid (B*H), block 256
__global__ __launch_bounds__(256) void kv_kernel(
    const float* __restrict__ V, const float* __restrict__ Ksm,
    float* __restrict__ kvT) {
  __shared__ float red[8 * 256];
  const int lane = threadIdx.x & 31;
  const int wave = threadIdx.x >> 5;
  const int half = lane >> 4;
  const int col = lane & 15;
  const int bh = blockIdx.x;
  const int b = bh / Hh, h = bh % Hh;

  const float* Vb = V + (size_t)b * Cc * Nn + (size_t)(h * DKk) * Nn;
  const float* Kh = Ksm + (size_t)h * Nn * DKk;

  v8f acc = ZERO8;
  const int nbase = wave * (Nn / 8);
#pragma unroll 4
  for (int k0 = nbase; k0 < nbase + Nn / 8; k0 += 4) {
    const int ka = k0 + 2 * half;
    v2f a;                                   // A[m=y][k=n] = V[y][ka..ka+1]
    a.x = Vb[(size_t)col * Nn + ka + 0];
    a.y = Vb[(size_t)col * Nn + ka + 1];
    v2f bf;                                  // B[k=n][x] = key[ka..ka+1][x]
    bf.x = Kh[(ka + 0) * DKk + col];
    bf.y = Kh[(ka + 1) * DKk + col];
    acc = wmma4(a, bf, acc);
  }
#pragma unroll
  for (int g = 0; g < 8; ++g)
    red[wave * 256 + (g + 8 * half) * 16 + col] = acc[g];
  __syncthreads();
  float s = 0.f;
#pragma unroll
  for (int w = 0; w < 8; ++w) s += red[w * 256 + threadIdx.x];
  kvT[(size_t)bh * 256 + threadIdx.x] = s;  // [y][x] linearized
}

// ---------------------------------------------------------------------------
// K4: XM[b, h*16+y, n] = sum_x kvT[y][x] * query[b, h*16+x, n] + V[...]
// (the +V term is attn_dyn, which equals value exactly)
// grid (N/128, B*H), block 256 (wave w -> n-tile w); kv tile staged in LDS.
__global__ __launch_bounds__(256) void attn_kernel(
    const float* __restrict__ Q, const float* __restrict__ V,
    const float* __restrict__ kvT, float* __restrict__ XM) {
  __shared__ float skv[256];
  const int lane = threadIdx.x & 31;
  const int wave = threadIdx.x >> 5;
  const int half = lane >> 4;
  const int col = lane & 15;
  const int bh = blockIdx.y;
  const int b = bh / Hh, h = bh % Hh;
  const int n0 = (blockIdx.x * 8 + wave) * 16;

  skv[threadIdx.x] = kvT[(size_t)bh * 256 + threadIdx.x];
  __syncthreads();

  const float* Qb = Q + (size_t)b * Cc * Nn + (size_t)(h * DKk) * Nn;

  v8f acc = ZERO8;
#pragma unroll
  for (int k0 = 0; k0 < DKk; k0 += 4) {
    const int ka = k0 + 2 * half;
    v2f a;                                   // A[m=y][k=x] = kvT[y][ka..ka+1]
    a.x = skv[col * 16 + ka + 0];
    a.y = skv[col * 16 + ka + 1];
    v2f bf;                                  // B[k=x][p] = Q[ka..ka+1][n0+col]
    bf.x = Qb[(size_t)(ka + 0) * Nn + n0 + col];
    bf.y = Qb[(size_t)(ka + 1) * Nn + n0 + col];
    acc = wmma4(a, bf, acc);
  }
#pragma unroll
  for (int g = 0; g < 8; ++g) {
    const int c = h * DKk + g + 8 * half;
    const size_t idx = (size_t)b * Cc * Nn + (size_t)c * Nn + n0 + col;
    XM[idx] = acc[g] + V[idx];  // attn_qkv + attn_dyn(=value)
  }
}

// ---------------------------------------------------------------------------
// K5: y = relu(Wc xm + bc); X1 = y*(saw+1) + sab + input
// grid (N/16, B), block 256
__global__ __launch_bounds__(256) void conv_c_kernel(
    const float* __restrict__ XM, const float* __restrict__ Wct,
    const float* __restrict__ bc, const float* __restrict__ saw,
    const float* __restrict__ sab, const float* __restrict__ Xin,
    float* __restrict__ X1) {
  __shared__ float xs[Cc * 16];
  const int lane = threadIdx.x & 31;
  const int wave = threadIdx.x >> 5;
  const int half = lane >> 4;
  const int col = lane & 15;
  const int n0 = blockIdx.x * 16;
  const int b = blockIdx.y;
  const int otile = wave * 16;

  const float* Xb = XM + (size_t)b * Cc * Nn;
  stage_tile(Xb, n0, xs);

  v8f acc = ZERO8, dummy = ZERO8;
  conv_kloop<false>(xs, Wct, nullptr, otile, half, col, acc, dummy);

#pragma unroll
  for (int g = 0; g < 8; ++g) {
    const int o = otile + g + 8 * half;
    const size_t sidx = (size_t)o * Nn + n0 + col;
    const size_t idx = (size_t)b * Cc * Nn + sidx;
    const float r = fmaxf(acc[g] + bc[o], 0.f);
    X1[idx] = r * (saw[sidx] + 1.f) + sab[sidx] + Xin[idx];
  }
}

// ---------------------------------------------------------------------------
// LayerNorm over (C,N) per batch: two-pass with atomic partials.
__global__ void zero_red_kernel(float* __restrict__ red) {
  if (threadIdx.x < 64) red[threadIdx.x] = 0.f;
}

__global__ __launch_bounds__(256) void ln_reduce_kernel(
    const float* __restrict__ X, float* __restrict__ red) {
  __shared__ float sh0[256], sh1[256];
  const int b = blockIdx.y;
  const size_t base =
      (size_t)b * Cc * Nn + (size_t)blockIdx.x * 4096 + threadIdx.x;
  float s = 0.f, s2 = 0.f;
#pragma unroll
  for (int i = 0; i < 16; ++i) {
    const float v = X[base + (size_t)i * 256];
    s += v;
    s2 += v * v;
  }
  sh0[threadIdx.x] = s;
  sh1[threadIdx.x] = s2;
  __syncthreads();
  for (int off = 128; off > 0; off >>= 1) {
    if (threadIdx.x < (unsigned)off) {
      sh0[threadIdx.x] += sh0[threadIdx.x + off];
      sh1[threadIdx.x] += sh1[threadIdx.x + off];
    }
    __syncthreads();
  }
  if (threadIdx.x == 0) {
    atomicAdd(&red[2 * b + 0], sh0[0]);
    atomicAdd(&red[2 * b + 1], sh1[0]);
  }
}

__global__ __launch_bounds__(256) void ln_apply_kernel(
    const float* __restrict__ X, const float* __restrict__ red,
    float* __restrict__ Y) {
  const int b = blockIdx.y;
  const float norm = 1.f / (float)(Cc * Nn);
  const float mu = red[2 * b + 0] * norm;
  const float ex2 = red[2 * b + 1] * norm;
  const float inv = rsqrtf(ex2 - mu * mu + LN_EPS);
  const size_t base =
      (size_t)b * Cc * Nn + (size_t)blockIdx.x * 1024 + threadIdx.x;
#pragma unroll
  for (int i = 0; i < 4; ++i) {
    const size_t idx = base + (size_t)i * 256;
    Y[idx] = (X[idx] - mu) * inv;
  }
}

// ---------------------------------------------------------------------------
// K7: GLU halves: GP = (Wg1 xn + bg1) * sigmoid(Wg2 xn + bg2)
// grid (N/16, B), block 256
__global__ __launch_bounds__(256) void glu12_kernel(
    const float* __restrict__ XN, const float* __restrict__ W1t,
    const float* __restrict__ W2t, const float* __restrict__ bg1,
    const float* __restrict__ bg2, float* __restrict__ GP) {
  __shared__ float xs[Cc * 16];
  const int lane = threadIdx.x & 31;
  const int wave = threadIdx.x >> 5;
  const int half = lane >> 4;
  const int col = lane & 15;
  const int n0 = blockIdx.x * 16;
  const int b = blockIdx.y;
  const int otile = wave * 16;

  const float* Xb = XN + (size_t)b * Cc * Nn;
  stage_tile(Xb, n0, xs);

  v8f acc1 = ZERO8, acc2 = ZERO8;
  conv_kloop<true>(xs, W1t, W2t, otile, half, col, acc1, acc2);

#pragma unroll
  for (int g = 0; g < 8; ++g) {
    const int o = otile + g + 8 * half;
    const size_t idx = (size_t)b * Cc * Nn + (size_t)o * Nn + n0 + col;
    const float a = acc1[g] + bg1[o];
    const float s = acc2[g] + bg2[o];
    GP[idx] = a * (1.f / (1.f + expf(-s)));
  }
}

// ---------------------------------------------------------------------------
// K8: X2 = (Wg3 gp + bg3 + xn) * (saw+1) + sab
// grid (N/16, B), block 256
__global__ __launch_bounds__(256) void glu3_kernel(
    const float* __restrict__ GP, const float* __restrict__ W3t,
    const float* __restrict__ bg3, const float* __restrict__ XN,
    const float* __restrict__ saw, const float* __restrict__ sab,
    float* __restrict__ X2) {
  __shared__ float xs[Cc * 16];
  const int lane = threadIdx.x & 31;
  const int wave = threadIdx.x >> 5;
  const int half = lane >> 4;
  const int col = lane & 15;
  const int n0 = blockIdx.x * 16;
  const int b = blockIdx.y;
  const int otile = wave * 16;

  const float* Xb = GP + (size_t)b * Cc * Nn;
  stage_tile(Xb, n0, xs);

  v8f acc = ZERO8, dummy = ZERO8;
  conv_kloop<false>(xs, W3t, nullptr, otile, half, col, acc, dummy);

#pragma unroll
  for (int g = 0; g < 8; ++g) {
    const int o = otile + g + 8 * half;
    const size_t sidx = (size_t)o * Nn + n0 + col;
    const size_t idx = (size_t)b * Cc * Nn + sidx;
    const float t = acc[g] + bg3[o] + XN[idx];
    X2[idx] = t * (saw[sidx] + 1.f) + sab[sidx];
  }
}

// ---------------------------------------------------------------------------
extern "C" void kernel_launch(void* const* d_in, const int* in_sizes, int n_in,
                              void* d_out, int out_size, void* d_ws,
                              size_t ws_size, hipStream_t stream) {
  const float* input = (const float*)d_in[0];
  const float* Wq = (const float*)d_in[1];
  const float* bq = (const float*)d_in[2];
  const float* Wv = (const float*)d_in[3];
  const float* bv = (const float*)d_in[4];
  const float* Wc = (const float*)d_in[5];
  const float* bc = (const float*)d_in[6];
  const float* memory = (const float*)d_in[7];
  const float* saw = (const float*)d_in[8];
  const float* sab = (const float*)d_in[9];
  // d_in[10], d_in[11] (nodevec1/2) unused: Aapt softmax row-sums == 1.
  const float* Wg1 = (const float*)d_in[12];
  const float* bg1 = (const float*)d_in[13];
  const float* Wg2 = (const float*)d_in[14];
  const float* bg2 = (const float*)d_in[15];
  const float* Wg3 = (const float*)d_in[16];
  const float* bg3 = (const float*)d_in[17];
  float* out = (float*)d_out;

  // Workspace layout (floats). Total ~25.6M floats (~102.4 MB).
  float* ws = (float*)d_ws;
  float* Wt = ws;                         // 6 * 16384
  float* Ksm = Wt + 6 * Cc * Cc;          // H*N*DK = 262144
  float* kvT = Ksm + Hh * Nn * DKk;       // B*H*256 = 65536
  float* red = kvT + Bb * Hh * 256;       // 64
  const size_t S = (size_t)Bb * Cc * Nn;  // 8388608
  float* buf0 = red + 64;
  float* buf1 = buf0 + S;
  float* buf2 = buf1 + S;

  const float* Wqt = Wt + 0 * Cc * Cc;
  const float* Wvt = Wt + 1 * Cc * Cc;
  const float* Wct = Wt + 2 * Cc * Cc;
  const float* W1t = Wt + 3 * Cc * Cc;
  const float* W2t = Wt + 4 * Cc * Cc;
  const float* W3t = Wt + 5 * Cc * Cc;

  transpose_w_kernel<<<dim3(64, 6), 256, 0, stream>>>(Wq, Wv, Wc, Wg1, Wg2,
                                                      Wg3, Wt);
  key_softmax_kernel<<<dim3(64), 256, 0, stream>>>(memory, Ksm);

  conv_qv_kernel<<<dim3(Nn / 16, Bb), 256, 0, stream>>>(input, Wqt, Wvt, bq,
                                                        bv, buf0, buf1);
  kv_kernel<<<dim3(Bb * Hh), 256, 0, stream>>>(buf1, Ksm, kvT);
  attn_kernel<<<dim3(Nn / 128, Bb * Hh), 256, 0, stream>>>(buf0, buf1, kvT,
                                                           buf2);
  conv_c_kernel<<<dim3(Nn / 16, Bb), 256, 0, stream>>>(buf2, Wct, bc, saw, sab,
                                                       input, buf0);

  zero_red_kernel<<<1, 64, 0, stream>>>(red);
  ln_reduce_kernel<<<dim3(64, Bb), 256, 0, stream>>>(buf0, red);
  ln_apply_kernel<<<dim3(256, Bb), 256, 0, stream>>>(buf0, red, buf1);

  glu12_kernel<<<dim3(Nn / 16, Bb), 256, 0, stream>>>(buf1, W1t, W2t, bg1, bg2,
                                                      buf2);
  glu3_kernel<<<dim3(Nn / 16, Bb), 256, 0, stream>>>(buf2, W3t, bg3, buf1, saw,
                                                     sab, buf0);

  zero_red_kernel<<<1, 64, 0, stream>>>(red);
  ln_reduce_kernel<<<dim3(64, Bb), 256, 0, stream>>>(buf0, red);
  ln_apply_kernel<<<dim3(256, Bb), 256, 0, stream>>>(buf0, red, out);
}